// HausdorffLoss_6468220748237
// MI455X (gfx1250) — compile-verified
//
#include <hip/hip_runtime.h>
#include <stdint.h>

#define BB   4
#define HH   256
#define WW   256
#define NTOT (BB * HH * WW)          /* 262144 */
#define FINF 1.0e6f

typedef __attribute__((ext_vector_type(2))) float v2f;
typedef __attribute__((ext_vector_type(8))) float v8f;

// ---------------------------------------------------------------------------
// Full-wave (32-lane) sum via V_WMMA_F32_16X16X4_F32 with a ones B-matrix.
// A (16x4 f32): lanes 0-15 hold K=0,1 (rows M=lane); lanes 16-31 hold K=2,3
// (rows M=lane-16). With a = {val, 0}: rowsum_m = val_m + val_{m+16} and
// D[m,n] = rowsum_m for every n. Lane l<16 holds D rows 0..7 of column l,
// lane l>=16 rows 8..15, so per-lane sum of the 8 D regs gives S_low /
// S_high; s + shfl_xor(s,16) = full 32-lane sum, uniform across the wave.
// Requires EXEC all ones (call only in uniform control flow).
// ---------------------------------------------------------------------------
__device__ __forceinline__ float wave_sum_wmma(float v) {
  v2f a;    a[0] = v;    a[1] = 0.0f;
  v2f ones; ones[0] = 1.0f; ones[1] = 1.0f;
  v8f c = {};
  v8f d = __builtin_amdgcn_wmma_f32_16x16x4_f32(
      /*neg_a=*/false, a, /*neg_b=*/false, ones,
      /*c_mod=*/(short)0, c, /*reuse_a=*/false, /*reuse_b=*/false);
  float s = 0.0f;
#pragma unroll
  for (int k = 0; k < 8; ++k) s += d[k];
  s += __shfl_xor(s, 16, 32);
  return s;
}

// CDNA5 async global->LDS copy of one dword per lane (ASYNCcnt-tracked).
__device__ __forceinline__ void async_ld_f32(float* lds_dst, const float* gsrc) {
  // Generic pointer to __shared__ carries the LDS byte offset in its low 32b.
  uint32_t lds_off = (uint32_t)(uintptr_t)lds_dst;
  asm volatile("global_load_async_to_lds_b32 %0, %1, off"
               :: "v"(lds_off), "v"(gsrc) : "memory");
}

// ---------------------------------------------------------------------------
// Kernel 1: vertical chamfer scans (phase 1 of exact EDT).
// One thread per (mask, b, j) column: g>>10 selects pred/true mask (uniform
// per wave -> no divergence), doubling parallelism vs fused scan.
// Lanes step rows together -> coalesced. Writes clamped vertical distance
// squared (d1^2), bit-matching the reference's cummin formulation.
// ---------------------------------------------------------------------------
__global__ __launch_bounds__(256)
void edt_phase1_kernel(const float* __restrict__ probs,
                       const float* __restrict__ targets,
                       float* __restrict__ d1p, float* __restrict__ d1t) {
  int g    = blockIdx.x * blockDim.x + threadIdx.x;   // 0 .. 2*B*W-1
  int m    = g >> 10;                                 // 0=pred, 1=true
  int col  = g & (BB * WW - 1);
  int b    = col >> 8;
  int j    = col & (WW - 1);
  int base = b * (HH * WW) + j;

  const float* src = m ? targets : probs;
  float        thr = m ? 0.5f    : 0.0f;   // sigmoid(x)>0.5 <=> x>0
  float*       dst = m ? d1t     : d1p;

  // forward scan (top -> bottom): running distance to nearest mask above
  float dv = FINF;
  for (int i = 0; i < HH; ++i) {
    int idx = base + i * WW;
    dv = (src[idx] > thr) ? 0.0f : dv + 1.0f;
    dst[idx] = fminf(dv, FINF);
  }
  // backward scan (bottom -> top), combine with forward, square
  dv = FINF;
  for (int i = HH - 1; i >= 0; --i) {
    int idx = base + i * WW;
    dv = (src[idx] > thr) ? 0.0f : dv + 1.0f;
    float a = fminf(dst[idx], fminf(dv, FINF));
    dst[idx] = a * a;                      // d1^2 in f32, matches reference
  }
}

// ---------------------------------------------------------------------------
// Kernel 2: exact lower envelope along W (phase 2) + loss, one block per row.
// D2[j] = min_j' d1sq[j'] + (j-j')^2 ; loss_j = (p-t)^2 * (D2p + D2t).
// Rows staged to LDS via the async-copy engine; inner loop reads LDS as
// float4 (ds_load_b128 broadcast) so the envelope is VALU-bound (fma+min).
// fmaf(diff,diff,d1sq) is bit-identical to square-then-add: diff^2 is an
// exact integer <= 255^2, so only the add rounds.
// ---------------------------------------------------------------------------
__global__ __launch_bounds__(256)
void edt_phase2_loss_kernel(const float* __restrict__ probs,
                            const float* __restrict__ targets,
                            const float* __restrict__ d1p,
                            const float* __restrict__ d1t,
                            float* __restrict__ partial) {
  __shared__ __align__(16) float shp[WW];
  __shared__ __align__(16) float sht[WW];
  __shared__ float swave[8];

  int row = blockIdx.x;           // b*H + i  (0 .. 1023)
  int j   = threadIdx.x;          // 0 .. 255
  int idx = row * WW + j;

  __builtin_prefetch(probs + idx, 0, 0);    // global_prefetch path
  __builtin_prefetch(targets + idx, 0, 0);

  // Stage both d1^2 rows into LDS with the CDNA5 async-copy engine.
  async_ld_f32(&shp[j], d1p + idx);
  async_ld_f32(&sht[j], d1t + idx);
  asm volatile("s_wait_asynccnt 0" ::: "memory");
  __syncthreads();

  const float4* shp4 = (const float4*)shp;
  const float4* sht4 = (const float4*)sht;

  float fj = (float)j;
  float Dp = 1.0e30f, Dt = 1.0e30f;
#pragma unroll 4
  for (int k4 = 0; k4 < WW / 4; ++k4) {     // LDS b128 broadcast reads
    float4 vp = shp4[k4];
    float4 vt = sht4[k4];
    float e0 = fj - (float)(4 * k4);        // exact integer diffs
    float e1 = e0 - 1.0f;
    float e2 = e0 - 2.0f;
    float e3 = e0 - 3.0f;
    Dp = fminf(Dp, fmaf(e0, e0, vp.x));
    Dp = fminf(Dp, fmaf(e1, e1, vp.y));
    Dp = fminf(Dp, fmaf(e2, e2, vp.z));
    Dp = fminf(Dp, fmaf(e3, e3, vp.w));
    Dt = fminf(Dt, fmaf(e0, e0, vt.x));
    Dt = fminf(Dt, fmaf(e1, e1, vt.y));
    Dt = fminf(Dt, fmaf(e2, e2, vt.z));
    Dt = fminf(Dt, fmaf(e3, e3, vt.w));
  }

  float x = probs[idx];
  float t = targets[idx];
  float p = 1.0f / (1.0f + __expf(-x));
  float e = p - t;
  float lossj = e * e * (Dp + Dt);

  // Wave reduction via the matrix engine (uniform flow, EXEC all ones).
  float s = wave_sum_wmma(lossj);
  if ((threadIdx.x & 31) == 0) swave[threadIdx.x >> 5] = s;
  __syncthreads();
  if (threadIdx.x == 0) {
    float tot = 0.0f;
#pragma unroll
    for (int w = 0; w < 8; ++w) tot += swave[w];    // fixed order
    partial[row] = tot;
  }
}

// ---------------------------------------------------------------------------
// Kernel 3: deterministic final reduction of 1024 block partials -> mean.
// ---------------------------------------------------------------------------
__global__ __launch_bounds__(256)
void final_reduce_kernel(const float* __restrict__ partial,
                         float* __restrict__ out) {
  __shared__ float swave[8];
  int t = threadIdx.x;
  float v = (partial[t] + partial[t + 256]) +
            (partial[t + 512] + partial[t + 768]);   // fixed order
  float s = wave_sum_wmma(v);
  if ((t & 31) == 0) swave[t >> 5] = s;
  __syncthreads();
  if (t == 0) {
    float tot = 0.0f;
#pragma unroll
    for (int w = 0; w < 8; ++w) tot += swave[w];
    out[0] = tot * (1.0f / (float)NTOT);
  }
}

// ---------------------------------------------------------------------------
extern "C" void kernel_launch(void* const* d_in, const int* in_sizes, int n_in,
                              void* d_out, int out_size, void* d_ws, size_t ws_size,
                              hipStream_t stream) {
  (void)in_sizes; (void)n_in; (void)out_size; (void)ws_size;
  const float* probs   = (const float*)d_in[0];
  const float* targets = (const float*)d_in[1];
  float* out = (float*)d_out;

  // Workspace layout: d1p [NTOT] f32 | d1t [NTOT] f32 | partial [B*H] f32
  float* d1p     = (float*)d_ws;
  float* d1t     = d1p + NTOT;
  float* partial = d1t + NTOT;

  edt_phase1_kernel<<<(2 * BB * WW) / 256, 256, 0, stream>>>(probs, targets, d1p, d1t);
  edt_phase2_loss_kernel<<<BB * HH, 256, 0, stream>>>(probs, targets, d1p, d1t, partial);
  final_reduce_kernel<<<1, 256, 0, stream>>>(partial, out);
}